// GraphSAGEModel_57767310131742
// MI455X (gfx1250) — compile-verified
//
#include <hip/hip_runtime.h>

#define FDIM 128
#define NN 50000
#define NE 600000
#define NG 512

typedef __attribute__((ext_vector_type(2))) float v2f;
typedef __attribute__((ext_vector_type(8))) float v8f;

__device__ __forceinline__ void atomAddF(float* p, float v) {
  __hip_atomic_fetch_add(p, v, __ATOMIC_RELAXED, __HIP_MEMORY_SCOPE_AGENT);
}

__global__ void zero_f32(float* __restrict__ p, int n4) {
  int i = blockIdx.x * blockDim.x + threadIdx.x;
  if (i < n4) ((float4*)p)[i] = make_float4(0.f, 0.f, 0.f, 0.f);
}

__global__ void deg_count(const int* __restrict__ dst, float* __restrict__ deg, int ne) {
  int e = blockIdx.x * blockDim.x + threadIdx.x;
  if (e < ne) atomAddF(&deg[dst[e]], 1.0f);
}

__global__ void batch_count(const int* __restrict__ batch, float* __restrict__ gcnt, int n) {
  int i = blockIdx.x * blockDim.x + threadIdx.x;
  if (i < n) atomAddF(&gcnt[batch[i]], 1.0f);
}

__global__ void deg_invert(float* __restrict__ deg, int n) {
  int i = blockIdx.x * blockDim.x + threadIdx.x;
  if (i < n) deg[i] = 1.0f / fmaxf(deg[i], 1.0f);
}

// One wave32 per edge: coalesced 512B row gather + 512B of contiguous f32 atomics.
__global__ void scatter_sum(const int* __restrict__ src, const int* __restrict__ dst,
                            const float* __restrict__ x, float* __restrict__ agg, int ne) {
  int gid = blockIdx.x * blockDim.x + threadIdx.x;
  int e = gid >> 5;
  int lane = gid & 31;
  if (e >= ne) return;
  int s = src[e];
  int d = dst[e];
  float4 v = ((const float4*)(x + (size_t)s * FDIM))[lane];
  float* ap = agg + (size_t)d * FDIM + lane * 4;
  atomAddF(ap + 0, v.x);
  atomAddF(ap + 1, v.y);
  atomAddF(ap + 2, v.z);
  atomAddF(ap + 3, v.w);
}

// Stage one 128x128 weight into LDS in WMMA-fragment-ready layout:
//   sW[kg*512 + n*4 + j] = W[(kg*4 + j)*128 + n],  kg=0..31, n=0..127, j=0..3
// Each lane's B fragment (k = kg*4 + 2*half, +1) is then one aligned 8-byte chunk
// -> single ds_load_b64 straight into a consecutive VGPR pair (no repack movs).
// Global reads stay coalesced (n varies across lanes).
__device__ __forceinline__ void stage_w_frag(float* __restrict__ sW,
                                             const float* __restrict__ W, int tid) {
  for (int p = tid; p < 32 * FDIM; p += 256) {
    int kg = p >> 7;
    int n = p & 127;
    int k0 = kg * 4;
    float4 v;
    v.x = W[(k0 + 0) * FDIM + n];
    v.y = W[(k0 + 1) * FDIM + n];
    v.z = W[(k0 + 2) * FDIM + n];
    v.w = W[(k0 + 3) * FDIM + n];
    ((float4*)sW)[kg * FDIM + n] = v;
  }
}

// One K-half (K=128) of the 16x128 output tile: 32 k-groups x 8 column tiles of
// v_wmma_f32_16x16x4_f32.  A fragment: lanes 0-15 -> K=k0,k0+1; lanes 16-31 -> K=k0+2,k0+3.
__device__ __forceinline__ void mm_half(const float* __restrict__ arow, float scale,
                                        const float* __restrict__ sW, int half, int l16,
                                        v8f acc[8]) {
  for (int kg = 0; kg < 32; kg++) {
    int k0 = kg * 4;
    v2f a;
    a.x = arow[k0] * scale;
    a.y = arow[k0 + 1] * scale;
    const float* bp = sW + kg * 512 + l16 * 4 + half * 2;
    #pragma unroll
    for (int t = 0; t < 8; t++) {
      v2f b = *(const v2f*)(bp + t * 64);
      acc[t] = __builtin_amdgcn_wmma_f32_16x16x4_f32(false, a, false, b, (short)0,
                                                     acc[t], false, false);
    }
  }
}

// out = act( (agg .* dinv) @ Wl + x @ Wr + bias ), 128x128 weights, fp32 WMMA.
// Block = 256 threads (8 wave32) -> 128 rows x 128 cols; weights staged in 64KB LDS,
// one 128-K half at a time.
template <int RELU>
__global__ void sage_gemm(const float* __restrict__ agg, const float* __restrict__ dinvv,
                          const float* __restrict__ xin, const float* __restrict__ Wl,
                          const float* __restrict__ Wr, const float* __restrict__ bias,
                          float* __restrict__ out, int nrows) {
  extern __shared__ float sW[];  // 128*128 floats = 64KB
  int tid = threadIdx.x;
  int wave = tid >> 5;
  int lane = tid & 31;
  int half = lane >> 4;
  int l16 = lane & 15;
  int row0 = blockIdx.x * 128 + wave * 16;
  bool active = (row0 < nrows);
  int row = row0 + l16;

  v8f acc[8];
  #pragma unroll
  for (int t = 0; t < 8; t++)
    #pragma unroll
    for (int r = 0; r < 8; r++) acc[t][r] = 0.0f;

  float dinv = active ? dinvv[row] : 0.0f;
  const float* arow = agg + (size_t)(active ? row : 0) * FDIM + half * 2;
  const float* xrow = xin + (size_t)(active ? row : 0) * FDIM + half * 2;

  stage_w_frag(sW, Wl, tid);
  __syncthreads();
  if (active) mm_half(arow, dinv, sW, half, l16, acc);
  __syncthreads();
  stage_w_frag(sW, Wr, tid);
  __syncthreads();
  if (active) {
    mm_half(xrow, 1.0f, sW, half, l16, acc);
    // Epilogue: C/D layout -> lanes 0-15: M=r, lanes 16-31: M=8+r, N=l16 within tile.
    #pragma unroll
    for (int t = 0; t < 8; t++) {
      int n = t * 16 + l16;
      float bv = bias[n];
      #pragma unroll
      for (int r = 0; r < 8; r++) {
        int m = row0 + half * 8 + r;
        float v = acc[t][r] + bv;
        if (RELU) v = fmaxf(v, 0.0f);
        out[(size_t)m * FDIM + n] = v;
      }
    }
  }
}

// Final layer (F_out=2) fused with global mean pool accumulation.
// One wave32 per node: float4 loads, shuffle reduction, 2 atomics.
__global__ void final_pool(const float* __restrict__ agg, const float* __restrict__ dinvv,
                           const float* __restrict__ h, const float* __restrict__ W4l,
                           const float* __restrict__ W4r, const float* __restrict__ b4,
                           const int* __restrict__ batch, float* __restrict__ gsum,
                           int nrows) {
  int gid = blockIdx.x * blockDim.x + threadIdx.x;
  int node = gid >> 5;
  int lane = gid & 31;
  if (node >= nrows) return;
  float dinv = dinvv[node];
  float4 a = ((const float4*)(agg + (size_t)node * FDIM))[lane];
  float4 xv = ((const float4*)(h + (size_t)node * FDIM))[lane];
  int k = lane * 4;
  float av[4] = {a.x, a.y, a.z, a.w};
  float xx[4] = {xv.x, xv.y, xv.z, xv.w};
  float acc0 = 0.f, acc1 = 0.f;
  #pragma unroll
  for (int j = 0; j < 4; j++) {
    float aj = av[j] * dinv;
    acc0 += aj * W4l[(k + j) * 2 + 0] + xx[j] * W4r[(k + j) * 2 + 0];
    acc1 += aj * W4l[(k + j) * 2 + 1] + xx[j] * W4r[(k + j) * 2 + 1];
  }
  #pragma unroll
  for (int off = 16; off > 0; off >>= 1) {
    acc0 += __shfl_xor(acc0, off, 32);
    acc1 += __shfl_xor(acc1, off, 32);
  }
  if (lane == 0) {
    int g = batch[node];
    atomAddF(&gsum[g * 2 + 0], acc0 + b4[0]);
    atomAddF(&gsum[g * 2 + 1], acc1 + b4[1]);
  }
}

__global__ void pool_finalize(const float* __restrict__ gsum, const float* __restrict__ gcnt,
                              float* __restrict__ out, int ngraphs) {
  int g = blockIdx.x * blockDim.x + threadIdx.x;
  if (g < ngraphs) {
    float c = fmaxf(gcnt[g], 1.0f);
    out[g * 2 + 0] = gsum[g * 2 + 0] / c;
    out[g * 2 + 1] = gsum[g * 2 + 1] / c;
  }
}

extern "C" void kernel_launch(void* const* d_in, const int* in_sizes, int n_in,
                              void* d_out, int out_size, void* d_ws, size_t ws_size,
                              hipStream_t stream) {
  const float* x = (const float*)d_in[0];
  const int* edge = (const int*)d_in[1];
  const int* batch = (const int*)d_in[2];
  const float* W1l = (const float*)d_in[3];
  const float* W1r = (const float*)d_in[4];
  const float* b1 = (const float*)d_in[5];
  const float* W2l = (const float*)d_in[6];
  const float* W2r = (const float*)d_in[7];
  const float* b2 = (const float*)d_in[8];
  const float* W3l = (const float*)d_in[9];
  const float* W3r = (const float*)d_in[10];
  const float* b3 = (const float*)d_in[11];
  const float* W4l = (const float*)d_in[12];
  const float* W4r = (const float*)d_in[13];
  const float* b4 = (const float*)d_in[14];
  float* out = (float*)d_out;

  float* ws = (float*)d_ws;
  float* deg = ws;                 // 50000
  float* gcnt = ws + 50000;        // 512
  float* gsum = ws + 50512;        // 1024
  float* agg = ws + 51536;         // 6.4M
  float* hA = agg + (size_t)NN * FDIM;
  float* hB = hA + (size_t)NN * FDIM;

  const int* src = edge;
  const int* dst = edge + NE;

  const int headN4 = 51536 / 4;
  const int aggN4 = (NN * FDIM) / 4;
  dim3 blk(256);
  dim3 zhead((headN4 + 255) / 256), zagg((aggN4 + 255) / 256);
  dim3 edgeG((NE + 255) / 256), nodeG((NN + 255) / 256);
  dim3 scatG(((size_t)NE * 32 + 255) / 256);
  dim3 gemmG((NN + 127) / 128);
  dim3 poolG(((size_t)NN * 32 + 255) / 256);
  size_t ldsW = (size_t)FDIM * FDIM * sizeof(float);  // 64KB

  // Graph-structure precompute (shared across all 4 layers).
  zero_f32<<<zhead, blk, 0, stream>>>(ws, headN4);
  deg_count<<<edgeG, blk, 0, stream>>>(dst, deg, NE);
  batch_count<<<nodeG, blk, 0, stream>>>(batch, gcnt, NN);
  deg_invert<<<nodeG, blk, 0, stream>>>(deg, NN);

  // Layer 1
  zero_f32<<<zagg, blk, 0, stream>>>(agg, aggN4);
  scatter_sum<<<scatG, blk, 0, stream>>>(src, dst, x, agg, NE);
  sage_gemm<1><<<gemmG, blk, ldsW, stream>>>(agg, deg, x, W1l, W1r, b1, hA, NN);
  // Layer 2
  zero_f32<<<zagg, blk, 0, stream>>>(agg, aggN4);
  scatter_sum<<<scatG, blk, 0, stream>>>(src, dst, hA, agg, NE);
  sage_gemm<1><<<gemmG, blk, ldsW, stream>>>(agg, deg, hA, W2l, W2r, b2, hB, NN);
  // Layer 3
  zero_f32<<<zagg, blk, 0, stream>>>(agg, aggN4);
  scatter_sum<<<scatG, blk, 0, stream>>>(src, dst, hB, agg, NE);
  sage_gemm<1><<<gemmG, blk, ldsW, stream>>>(agg, deg, hB, W3l, W3r, b3, hA, NN);
  // Layer 4 fused with pooling
  zero_f32<<<zagg, blk, 0, stream>>>(agg, aggN4);
  scatter_sum<<<scatG, blk, 0, stream>>>(src, dst, hA, agg, NE);
  final_pool<<<poolG, blk, 0, stream>>>(agg, deg, hA, W4l, W4r, b4, batch, gsum, NN);
  pool_finalize<<<(NG + 255) / 256, blk, 0, stream>>>(gsum, gcnt, out, NG);
}